// GridSamplePScan_64089501991430
// MI455X (gfx1250) — compile-verified
//
#include <hip/hip_runtime.h>

#define LL 32
#define CC 8
#define HH 64
#define WW 64
#define HWSZ 4096   // H*W

typedef float f4 __attribute__((ext_vector_type(4)));
typedef unsigned int u32x4 __attribute__((ext_vector_type(4)));
typedef unsigned int u32x8 __attribute__((ext_vector_type(8)));

// ---------------------------------------------------------------------------
// Transpose images (B,L,C,H,W) -> imgT (B,L,H,W,C) so each pixel's 8 channels
// are one contiguous 32B vector (2 x b128 per bilinear corner).
// ---------------------------------------------------------------------------
__global__ __launch_bounds__(256) void gsps_transpose_kernel(
    const float* __restrict__ images, float* __restrict__ imgT) {
  const int idx = blockIdx.x * 256 + threadIdx.x;  // (b*L + l)*HW + p
  const int p  = idx & (HWSZ - 1);
  const int bl = idx >> 12;
  const float* src = images + (size_t)bl * CC * HWSZ + p;
  float* dst = imgT + ((size_t)bl * HWSZ + p) * CC;
#pragma unroll
  for (int c = 0; c < CC; ++c) dst[c] = src[(size_t)c * HWSZ];
}

// ---------------------------------------------------------------------------
// out[b,t,c,y,x] = sum_{k<=t} bilinear(images[b,k,c], base + (cum[t]-cum[k]))
// cum[t]-cum[k] = sum_{j=k+1..t} flows[j]  -> running accumulator, k = t..0.
// grid = (16 pixel tiles of 4 rows, 32 t, 2 b), block = 256 (one pixel/thread).
// flows[b, 0..t, :, tile] staged into LDS by the Tensor Data Mover: one
// tensor_load_to_lds DMA (2-D tile: 2(t+1) rows of 256 floats, row stride
// 4096 floats), issued by wave 0 via inline asm, completion via TENSORcnt.
// TR=true: images are transposed (B,L,H,W,C); TR=false: original layout.
// ---------------------------------------------------------------------------
template <bool TR>
__global__ __launch_bounds__(256) void gsps_warp_scan_kernel(
    const float* __restrict__ images, const float* __restrict__ flows,
    float* __restrict__ out) {
  __shared__ float sflow[LL * 2 * 256];  // [k*2+comp][pixel-in-tile]

  const int tile = blockIdx.x;   // 0..15
  const int t    = blockIdx.y;   // 0..31
  const int b    = blockIdx.z;   // 0..1
  const int tid  = threadIdx.x;  // 0..255
  const int pix0 = tile * 256;

  const float* gbase = flows + (size_t)b * LL * 2 * HWSZ;

  // ---- Stage via Tensor Data Mover (one DMA, issued by wave 0) ----
  if (tid < 32) {
    const unsigned lds_base = (unsigned)(size_t)(&sflow[0]);
    const unsigned long long ga = (unsigned long long)(size_t)(gbase + pix0);

    u32x4 g0;
    g0.x = 1u;                                       // count=1 (valid, user mode)
    g0.y = lds_base;                                 // lds_addr (bytes)
    g0.z = (unsigned)(ga & 0xFFFFFFFFull);           // global_addr[31:0]
    g0.w = (unsigned)((ga >> 32) & 0x01FFFFFFull)    // global_addr[56:32]
           | 0x80000000u;                            // type = 2 ("image")

    u32x8 g1;
    g1[0] = 0x00020000u;                  // wg_mask=0, data_size=2 (4B), no pad
    g1[1] = 256u << 16;                   // atomic_bar_addr=0; tensor_dim0=256 lo16
    g1[2] = 64u << 16;                    // tensor_dim0 hi=0; tensor_dim1=64 lo16
    g1[3] = 256u << 16;                   // tensor_dim1 hi=0; tile_dim0=256
    g1[4] = 2u * (unsigned)(t + 1);       // tile_dim1=2(t+1); tile_dim2=0
    g1[5] = 4096u;                        // tensor_dim0_stride[31:0] (elements)
    g1[6] = 0u;                           // stride hi / tensor_dim1_stride lo
    g1[7] = 0u;

    const u32x4 g2 = {0u, 0u, 0u, 0u};    // tensor_dim2=0, tile_dim3=0 (unused)
    const u32x4 g3 = {0u, 0u, 0u, 0u};    // tensor_dim3_stride=0, tile_dim4=0

    asm volatile("tensor_load_to_lds %0, %1, %2, %3\n\t"
                 "s_wait_tensorcnt 0x0"
                 :
                 : "s"(g0), "s"(g1), "s"(g2), "s"(g3)
                 : "memory");
  }
  __syncthreads();

  // ---- Per-thread pixel ----
  const int x = tid & 63;
  const int y = (tile << 2) + (tid >> 6);

  f4 accLo = {0.f, 0.f, 0.f, 0.f};
  f4 accHi = {0.f, 0.f, 0.f, 0.f};
  float accS[CC];
#pragma unroll
  for (int c = 0; c < CC; ++c) accS[c] = 0.0f;

  float dx = 0.0f, dy = 0.0f;  // = cum[t] - cum[k], maintained incrementally

  for (int k = t; k >= 0; --k) {
    // ix = ((x + 0.5 + 32*dx) mod 64) - 0.5   (x wraps); iy = y + 32*dy
    float px = (float)x + 0.5f + dx * 32.0f;
    px = px - floorf(px * 0.015625f) * 64.0f;
    const float ix = px - 0.5f;
    const float iy = (float)y + dy * 32.0f;

    const float x0f = floorf(ix), y0f = floorf(iy);
    const float wx1 = ix - x0f,   wy1 = iy - y0f;
    const float wx0 = 1.0f - wx1, wy0 = 1.0f - wy1;
    const int x0 = (int)x0f, y0 = (int)y0f;
    const int x1 = x0 + 1,   y1 = y0 + 1;

    const float vx0 = (x0 >= 0 && x0 < WW) ? 1.0f : 0.0f;
    const float vx1 = (x1 >= 0 && x1 < WW) ? 1.0f : 0.0f;
    const float vy0 = (y0 >= 0 && y0 < HH) ? 1.0f : 0.0f;
    const float vy1 = (y1 >= 0 && y1 < HH) ? 1.0f : 0.0f;

    const float w00 = wx0 * wy0 * vx0 * vy0;
    const float w01 = wx1 * wy0 * vx1 * vy0;
    const float w10 = wx0 * wy1 * vx0 * vy1;
    const float w11 = wx1 * wy1 * vx1 * vy1;

    const int xi0 = min(max(x0, 0), WW - 1);
    const int xi1 = min(max(x1, 0), WW - 1);
    const int yi0 = min(max(y0, 0), HH - 1);
    const int yi1 = min(max(y1, 0), HH - 1);

    const int o00 = yi0 * WW + xi0;
    const int o01 = yi0 * WW + xi1;
    const int o10 = yi1 * WW + xi0;
    const int o11 = yi1 * WW + xi1;

    if (TR) {
      // (B,L,H,W,C): each corner = 2 contiguous f4 (32B), fully in-bounds
      const f4* ib = (const f4*)(images + ((size_t)(b * LL + k) * HWSZ) * CC);
      accLo += w00 * ib[o00 * 2 + 0] + w01 * ib[o01 * 2 + 0] +
               w10 * ib[o10 * 2 + 0] + w11 * ib[o11 * 2 + 0];
      accHi += w00 * ib[o00 * 2 + 1] + w01 * ib[o01 * 2 + 1] +
               w10 * ib[o10 * 2 + 1] + w11 * ib[o11 * 2 + 1];
    } else {
      const float* imgk = images + (size_t)(b * LL + k) * CC * HWSZ;
#pragma unroll
      for (int c = 0; c < CC; ++c) {
        const float* ic = imgk + c * HWSZ;
        accS[c] += w00 * ic[o00] + w01 * ic[o01] + w10 * ic[o10] + w11 * ic[o11];
      }
    }

    // advance displacement: d(k-1) = d(k) + flows[k]
    dx += sflow[(k * 2 + 0) * 256 + tid];
    dy += sflow[(k * 2 + 1) * 256 + tid];
  }

  float* op = out + (size_t)(b * LL + t) * CC * HWSZ + y * WW + x;
  if (TR) {
#pragma unroll
    for (int c = 0; c < 4; ++c) op[c * HWSZ] = accLo[c];
#pragma unroll
    for (int c = 0; c < 4; ++c) op[(c + 4) * HWSZ] = accHi[c];
  } else {
#pragma unroll
    for (int c = 0; c < CC; ++c) op[c * HWSZ] = accS[c];
  }
}

// ---------------------------------------------------------------------------
extern "C" void kernel_launch(void* const* d_in, const int* in_sizes, int n_in,
                              void* d_out, int out_size, void* d_ws, size_t ws_size,
                              hipStream_t stream) {
  const float* flows  = (const float*)d_in[0];  // (2,32,2,64,64) f32
  const float* images = (const float*)d_in[1];  // (2,32,8,64,64) f32
  float* out = (float*)d_out;                   // (2,32,8,64,64) f32

  const size_t needT = (size_t)2 * LL * CC * HWSZ * sizeof(float);  // 8 MiB
  dim3 grid(16, 32, 2);  // (pixel tile, t, b)

  if (ws_size >= needT) {
    float* imgT = (float*)d_ws;
    gsps_transpose_kernel<<<1024, 256, 0, stream>>>(images, imgT);
    gsps_warp_scan_kernel<true><<<grid, 256, 0, stream>>>(imgT, flows, out);
  } else {
    gsps_warp_scan_kernel<false><<<grid, 256, 0, stream>>>(images, flows, out);
  }
}